// GCN_24489903522241
// MI455X (gfx1250) — compile-verified
//
#include <hip/hip_runtime.h>
#include <hip/hip_bf16.h>

typedef __attribute__((ext_vector_type(2))) float v2f;
typedef __attribute__((ext_vector_type(8))) float v8f;

#define IN_F 28
#define HID  16
#define OUTC 2

__device__ __forceinline__ void atomic_add_f32(float* p, float v) {
    __hip_atomic_fetch_add(p, v, __ATOMIC_RELAXED, __HIP_MEMORY_SCOPE_AGENT);
}

// deg[i] = 1.0f  (self-loop contribution)
__global__ void k_init_deg(float* __restrict__ deg, int n) {
    int i = blockIdx.x * blockDim.x + threadIdx.x;
    if (i < n) deg[i] = 1.0f;
}

// deg[dst[e]] += 1 over real edges
__global__ void k_deg(const long long* __restrict__ dst, float* __restrict__ deg, int e) {
    int i = blockIdx.x * blockDim.x + threadIdx.x;
    if (i < e) atomic_add_f32(&deg[(int)dst[i]], 1.0f);
}

// dis[i] = rsqrt(deg[i])   (deg >= 1 always due to self-loops) -- in place
__global__ void k_dis(float* __restrict__ deg, int n) {
    int i = blockIdx.x * blockDim.x + threadIdx.x;
    if (i < n) deg[i] = rsqrtf(deg[i]);
}

// hs1[m, :] = (x[m, :] @ W1) * dis[m]   via V_WMMA_F32_16X16X4_F32
// One wave32 per 16-row tile; K=28 -> 7 chained WMMAs.
// A 16x4 f32 layout: lane = M + 16*(K>=2), vgpr = K&1 (ISA 7.12.2)
// B  4x16 f32 layout mirrors: vgpr0 = {K=0 lanes0-15, K=2 lanes16-31}, vgpr1 = {K=1, K=3}
__global__ void k_gemm1_wmma(const float* __restrict__ x, const float* __restrict__ W1,
                             const float* __restrict__ dis, float* __restrict__ hs1, int n) {
    int wave = threadIdx.x >> 5;
    int lane = threadIdx.x & 31;
    int tile = blockIdx.x * (blockDim.x >> 5) + wave;
    int base = tile * 16;
    if (base >= n) return;                      // wave-uniform guard, EXEC stays all-1 for WMMA
    int hi  = lane >> 4;                        // 0: K pair {0,1} / M 0..7 ; 1: K pair {2,3} / M 8..15
    int low = lane & 15;

    int m = base + low;
    if (m >= n) m = n - 1;                      // clamp loads; stores guarded below
    const float* xrow = x + (size_t)m * IN_F;

    v8f c = {};
#pragma unroll
    for (int kk = 0; kk < IN_F / 4; ++kk) {
        int k0 = kk * 4 + 2 * hi;
        v2f a; a.x = xrow[k0];              a.y = xrow[k0 + 1];
        v2f b; b.x = W1[k0 * HID + low];    b.y = W1[(k0 + 1) * HID + low];
        c = __builtin_amdgcn_wmma_f32_16x16x4_f32(
                /*neg_a=*/false, a, /*neg_b=*/false, b,
                /*c_mod=*/(short)0, c, /*reuse_a=*/false, /*reuse_b=*/false);
    }
    // C/D layout: vgpr r: M = base + r + 8*hi, N = low
    if (base + 16 <= n) {
        // full tile: branch-free epilogue (always taken for N % 16 == 0)
#pragma unroll
        for (int r = 0; r < 8; ++r) {
            int row = base + r + 8 * hi;
            hs1[(size_t)row * HID + low] = c[r] * dis[row];
        }
    } else {
#pragma unroll
        for (int r = 0; r < 8; ++r) {
            int row = base + r + 8 * hi;
            if (row < n) hs1[(size_t)row * HID + low] = c[r] * dis[row];
        }
    }
}

// agg1[dst] += hs1[src]  (16 floats/edge, pre-scaled by dis[src])
__global__ void k_scatter16(const long long* __restrict__ src, const long long* __restrict__ dst,
                            const float* __restrict__ hs1, float* __restrict__ agg1, int e) {
    int i = blockIdx.x * blockDim.x + threadIdx.x;
    if (i >= e) return;
    int s = (int)src[i];
    int d = (int)dst[i];
    const float4* hp = (const float4*)(hs1 + (size_t)s * 16);
    float4 v0 = hp[0], v1 = hp[1], v2 = hp[2], v3 = hp[3];
    float* ap = agg1 + (size_t)d * 16;
    atomic_add_f32(ap + 0,  v0.x); atomic_add_f32(ap + 1,  v0.y);
    atomic_add_f32(ap + 2,  v0.z); atomic_add_f32(ap + 3,  v0.w);
    atomic_add_f32(ap + 4,  v1.x); atomic_add_f32(ap + 5,  v1.y);
    atomic_add_f32(ap + 6,  v1.z); atomic_add_f32(ap + 7,  v1.w);
    atomic_add_f32(ap + 8,  v2.x); atomic_add_f32(ap + 9,  v2.y);
    atomic_add_f32(ap + 10, v2.z); atomic_add_f32(ap + 11, v2.w);
    atomic_add_f32(ap + 12, v3.x); atomic_add_f32(ap + 13, v3.y);
    atomic_add_f32(ap + 14, v3.z); atomic_add_f32(ap + 15, v3.w);
}

// Node epilogue of layer 1 + layer-2 transform:
// h2 = relu(dis*(agg1 + hs1) + b1) ; gs2 = (h2 @ W2) * dis
__global__ void k_node2(const float* __restrict__ agg1, const float* __restrict__ hs1,
                        const float* __restrict__ dis, const float* __restrict__ b1,
                        const float* __restrict__ W2, float* __restrict__ gs2, int n) {
    int i = blockIdx.x * blockDim.x + threadIdx.x;
    if (i >= n) return;
    float di = dis[i];
    float o0 = 0.f, o1 = 0.f;
    const float* a = agg1 + (size_t)i * 16;
    const float* h = hs1  + (size_t)i * 16;
#pragma unroll
    for (int j = 0; j < 16; ++j) {
        float hv = fmaf(di, a[j] + h[j], b1[j]);
        hv = fmaxf(hv, 0.f);
        o0 = fmaf(hv, W2[j * 2 + 0], o0);
        o1 = fmaf(hv, W2[j * 2 + 1], o1);
    }
    gs2[(size_t)i * 2 + 0] = o0 * di;
    gs2[(size_t)i * 2 + 1] = o1 * di;
}

// agg2[dst] += gs2[src]  (2 floats/edge)
__global__ void k_scatter2(const long long* __restrict__ src, const long long* __restrict__ dst,
                           const float* __restrict__ gs2, float* __restrict__ agg2, int e) {
    int i = blockIdx.x * blockDim.x + threadIdx.x;
    if (i >= e) return;
    int s = (int)src[i];
    int d = (int)dst[i];
    float2 v = *(const float2*)(gs2 + (size_t)s * 2);
    atomic_add_f32(agg2 + (size_t)d * 2 + 0, v.x);
    atomic_add_f32(agg2 + (size_t)d * 2 + 1, v.y);
}

// z = dis*(agg2 + gs2) + b2 ; out = log_softmax(z)
__global__ void k_out(const float* __restrict__ agg2, const float* __restrict__ gs2,
                      const float* __restrict__ dis, const float* __restrict__ b2,
                      float* __restrict__ out, int n) {
    int i = blockIdx.x * blockDim.x + threadIdx.x;
    if (i >= n) return;
    float di = dis[i];
    float z0 = fmaf(di, agg2[i * 2 + 0] + gs2[i * 2 + 0], b2[0]);
    float z1 = fmaf(di, agg2[i * 2 + 1] + gs2[i * 2 + 1], b2[1]);
    float m  = fmaxf(z0, z1);
    float l  = logf(__expf(z0 - m) + __expf(z1 - m));
    out[i * 2 + 0] = z0 - m - l;
    out[i * 2 + 1] = z1 - m - l;
}

extern "C" void kernel_launch(void* const* d_in, const int* in_sizes, int n_in,
                              void* d_out, int out_size, void* d_ws, size_t ws_size,
                              hipStream_t stream) {
    const float*     x   = (const float*)d_in[0];
    const long long* ei  = (const long long*)d_in[1];   // int64 edge_index [2, E]
    const float*     W1  = (const float*)d_in[2];
    const float*     b1  = (const float*)d_in[3];
    const float*     W2  = (const float*)d_in[4];
    const float*     b2  = (const float*)d_in[5];

    int n = in_sizes[0] / IN_F;         // 200000
    int e = in_sizes[1] / 2;            // 6400000
    const long long* src = ei;
    const long long* dst = ei + (size_t)e;

    // Workspace layout (floats): dis[n] | hs1[n*16] | agg1[n*16] | gs2[n*2] | agg2[n*2]
    float* ws   = (float*)d_ws;
    float* dis  = ws;
    float* hs1  = dis  + (size_t)n;
    float* agg1 = hs1  + (size_t)n * 16;
    float* gs2  = agg1 + (size_t)n * 16;
    float* agg2 = gs2  + (size_t)n * 2;

    hipMemsetAsync(agg1, 0, (size_t)n * 16 * sizeof(float), stream);
    hipMemsetAsync(agg2, 0, (size_t)n * 2  * sizeof(float), stream);

    const int B = 256;
    int gn = (n + B - 1) / B;
    int ge = (e + B - 1) / B;

    k_init_deg<<<gn, B, 0, stream>>>(dis, n);
    k_deg<<<ge, B, 0, stream>>>(dst, dis, e);
    k_dis<<<gn, B, 0, stream>>>(dis, n);

    int nTiles = (n + 15) / 16;                 // one wave32 per 16-row tile
    int wavesPerBlock = B / 32;                 // 8
    int gg = (nTiles + wavesPerBlock - 1) / wavesPerBlock;
    k_gemm1_wmma<<<gg, B, 0, stream>>>(x, W1, dis, hs1, n);

    k_scatter16<<<ge, B, 0, stream>>>(src, dst, hs1, agg1, e);
    k_node2<<<gn, B, 0, stream>>>(agg1, hs1, dis, b1, W2, gs2, n);
    k_scatter2<<<ge, B, 0, stream>>>(src, dst, gs2, agg2, e);
    k_out<<<gn, B, 0, stream>>>(agg2, gs2, dis, b2, (float*)d_out, n);
}